// GQNN_39994735461030
// MI455X (gfx1250) — compile-verified
//
#include <hip/hip_runtime.h>
#include <hip/hip_bf16.h>
#include <math.h>

typedef __attribute__((ext_vector_type(2))) float v2f;
typedef __attribute__((ext_vector_type(8))) float v8f;

#define DIM 128
#define BPAD 132   // padded K-stride for transposed B panels in LDS (4*132 % 256 -> conflict-free)

// Native CDNA5 fp32 atomic add (no-return form: tracked by STOREcnt, implicit wait at endpgm)
__device__ __forceinline__ void gatomic_add_f32(float* addr, float v) {
    asm volatile("global_atomic_add_f32 %0, %1, off" :: "v"(addr), "v"(v) : "memory");
}

// ---------------- zero fill ----------------
__global__ void zero_f32(float* __restrict__ p, int n) {
    int i = blockIdx.x * blockDim.x + threadIdx.x;
    int i4 = i * 4;
    if (i4 + 3 < n) {
        *(float4*)(p + i4) = make_float4(0.f, 0.f, 0.f, 0.f);
    } else {
        for (int j = i4; j < n; ++j) p[j] = 0.f;
    }
}

// ---------------- degree ----------------
__global__ void degree_kernel(const int* __restrict__ dst, float* __restrict__ deg, int n_edges) {
    int e = blockIdx.x * blockDim.x + threadIdx.x;
    if (e < n_edges) gatomic_add_f32(deg + dst[e], 1.0f);
}

// ---------------- edge gather + scatter-add (one wave32 per edge) ----------------
__global__ void scatter_kernel(const float* __restrict__ xin, const int* __restrict__ src,
                               const int* __restrict__ dst, float* __restrict__ agg,
                               int n_edges) {
    int wave = (blockIdx.x * blockDim.x + threadIdx.x) >> 5;
    int lane = threadIdx.x & 31;
    if (wave >= n_edges) return;
    int s = src[wave];
    int d = dst[wave];
    float4 v = ((const float4*)(xin + (size_t)s * DIM))[lane];
    float* arow = agg + (size_t)d * DIM + lane * 4;
    gatomic_add_f32(arow + 0, v.x);
    gatomic_add_f32(arow + 1, v.y);
    gatomic_add_f32(arow + 2, v.z);
    gatomic_add_f32(arow + 3, v.w);
}

// ---------------- fused SAGE layer GEMM ----------------
// out = relu( (agg/max(deg,1)) @ Wl + bl + xin @ Wr )   [n x 128] @ [128 x 128]
// Block = 8 waves sharing one 16-column weight panel staged in LDS;
// each wave computes one 16x16 output tile via V_WMMA_F32_16X16X4_F32.
__global__ void sage_gemm_kernel(const float* __restrict__ agg, const float* __restrict__ deg,
                                 const float* __restrict__ xin,
                                 const float* __restrict__ Wl, const float* __restrict__ bl,
                                 const float* __restrict__ Wr,
                                 float* __restrict__ out, int n_nodes) {
    __shared__ float Blt[16 * BPAD];   // Wl panel, transposed: Blt[n*BPAD + k]
    __shared__ float Brt[16 * BPAD];   // Wr panel, transposed

    int col_tile = blockIdx.x & 7;     // which 16-column slice of the 128 outputs
    int group    = blockIdx.x >> 3;    // group of 8 row tiles
    int wv       = threadIdx.x >> 5;
    int lane     = threadIdx.x & 31;
    int colbase  = col_tile * 16;

    // Cooperative load of both 128x16 panels (transposed into LDS, K contiguous per column)
    for (int idx = threadIdx.x; idx < 2048; idx += blockDim.x) {
        int k = idx >> 4, n = idx & 15;
        Blt[n * BPAD + k] = Wl[(size_t)k * DIM + colbase + n];
        Brt[n * BPAD + k] = Wr[(size_t)k * DIM + colbase + n];
    }
    __syncthreads();

    int row_tile = group * 8 + wv;
    if (row_tile * 16 + 16 > n_nodes) return;   // wave-uniform: EXEC stays all-ones

    int m  = lane & 15;            // A row held by this lane
    int kk = (lane >> 4) * 2;      // lanes 0-15: K={0,1}; lanes 16-31: K={2,3}
    int nn = lane & 15;            // B column held by this lane

    int row = row_tile * 16 + m;
    float rdeg = 1.0f / fmaxf(deg[row], 1.0f);
    const float* arow = agg + (size_t)row * DIM;
    const float* xrow = xin + (size_t)row * DIM;
    const float* blane = Blt + nn * BPAD;
    const float* rlane = Brt + nn * BPAD;

    v8f acc = {};
#pragma unroll 4
    for (int k0 = 0; k0 < DIM; k0 += 4) {
        // neighbor-mean term
        v2f a1 = *(const v2f*)(arow + k0 + kk);
        a1.x *= rdeg; a1.y *= rdeg;
        v2f b1 = *(const v2f*)(blane + k0 + kk);     // one conflict-free ds_load_b64
        acc = __builtin_amdgcn_wmma_f32_16x16x4_f32(false, a1, false, b1,
                                                    (short)0, acc, false, false);
        // root term
        v2f a2 = *(const v2f*)(xrow + k0 + kk);
        v2f b2 = *(const v2f*)(rlane + k0 + kk);
        acc = __builtin_amdgcn_wmma_f32_16x16x4_f32(false, a2, false, b2,
                                                    (short)0, acc, false, false);
    }

    // C/D layout: VGPR i -> rows i (lanes 0-15) and i+8 (lanes 16-31), col = lane&15
    int ccol  = colbase + (lane & 15);
    int rbase = row_tile * 16 + (lane >> 4) * 8;
    float bias = bl[ccol];
#pragma unroll
    for (int i = 0; i < 8; ++i) {
        float v = acc[i] + bias;
        out[(size_t)(rbase + i) * DIM + ccol] = fmaxf(v, 0.0f);
    }
}

// ---------------- heads: one wave32 per node ----------------
__global__ void head_kernel(const float* __restrict__ h,
                            const float* __restrict__ Wp, const float* __restrict__ bp,
                            const float* __restrict__ Wd, const float* __restrict__ bd,
                            float* __restrict__ out, int n_nodes) {
    int wave = (blockIdx.x * blockDim.x + threadIdx.x) >> 5;
    int lane = threadIdx.x & 31;
    if (wave >= n_nodes) return;
    float4 hv = ((const float4*)(h + (size_t)wave * DIM))[lane];
    float4 wp = ((const float4*)Wp)[lane];
    float4 wd = ((const float4*)Wd)[lane];
    float p = hv.x * wp.x + hv.y * wp.y + hv.z * wp.z + hv.w * wp.w;
    float d = hv.x * wd.x + hv.y * wd.y + hv.z * wd.z + hv.w * wd.w;
#pragma unroll
    for (int off = 16; off > 0; off >>= 1) {
        p += __shfl_xor(p, off, 32);
        d += __shfl_xor(d, off, 32);
    }
    if (lane == 0) {
        p += bp[0];
        d += bd[0];
        d = 1.0f / (1.0f + __expf(-d));
        out[wave] = p - d;                // preds - diffs
        out[n_nodes + wave] = p + d;      // preds + diffs
    }
}

extern "C" void kernel_launch(void* const* d_in, const int* in_sizes, int n_in,
                              void* d_out, int out_size, void* d_ws, size_t ws_size,
                              hipStream_t stream) {
    const float* x    = (const float*)d_in[0];
    const int*   edge = (const int*)d_in[1];
    const float* W1l  = (const float*)d_in[2];
    const float* b1l  = (const float*)d_in[3];
    const float* W1r  = (const float*)d_in[4];
    const float* W2l  = (const float*)d_in[5];
    const float* b2l  = (const float*)d_in[6];
    const float* W2r  = (const float*)d_in[7];
    const float* Wp   = (const float*)d_in[8];
    const float* bp   = (const float*)d_in[9];
    const float* Wd   = (const float*)d_in[10];
    const float* bd   = (const float*)d_in[11];
    float* out = (float*)d_out;

    const int N = in_sizes[0] / DIM;       // 100000
    const int E = in_sizes[1] / 2;         // 1600000
    const int* src = edge;                 // edge_index[0]
    const int* dst = edge + E;             // edge_index[1]

    // workspace layout
    char* ws = (char*)d_ws;
    float* agg = (float*)ws;                              // N*128 floats
    float* h1  = agg + (size_t)N * DIM;                   // N*128
    float* h2  = h1 + (size_t)N * DIM;                    // N*128
    float* deg = h2 + (size_t)N * DIM;                    // N

    const int BLK = 256;
    int nfeat = N * DIM;
    int zgrid = (nfeat / 4 + BLK - 1) / BLK;
    int dgrid = (N / 4 + BLK) / BLK;
    int egrid_t = (E + BLK - 1) / BLK;                    // 1 thread / edge
    int egrid_w = (E * 32 + BLK - 1) / BLK;               // 1 wave / edge
    int rowTiles = (N + 15) / 16;
    int ggrid = 8 * ((rowTiles + 7) / 8);                 // 8 col-tiles x row-groups of 8
    int hgrid = (N * 32 + BLK - 1) / BLK;

    // ---- layer 1 ----
    zero_f32<<<zgrid, BLK, 0, stream>>>(agg, nfeat);
    zero_f32<<<dgrid, BLK, 0, stream>>>(deg, N);
    degree_kernel<<<egrid_t, BLK, 0, stream>>>(dst, deg, E);
    scatter_kernel<<<egrid_w, BLK, 0, stream>>>(x, src, dst, agg, E);
    sage_gemm_kernel<<<ggrid, BLK, 0, stream>>>(agg, deg, x, W1l, b1l, W1r, h1, N);

    // ---- layer 2 ----
    zero_f32<<<zgrid, BLK, 0, stream>>>(agg, nfeat);
    scatter_kernel<<<egrid_w, BLK, 0, stream>>>(h1, src, dst, agg, E);
    sage_gemm_kernel<<<ggrid, BLK, 0, stream>>>(agg, deg, h1, W2l, b2l, W2r, h2, N);

    // ---- heads ----
    head_kernel<<<hgrid, BLK, 0, stream>>>(h2, Wp, bp, Wd, bd, out, N);
}